// GHMRLoss_79087527788873
// MI455X (gfx1250) — compile-verified
//
#include <hip/hip_runtime.h>
#include <hip/hip_bf16.h>

#define GHMR_BINS 10
#define GHMR_MIU  0.02f
// MIU*MIU folded at compile time in f32, matching reference's f32 miu*miu
#define GHMR_MIU2 (GHMR_MIU * GHMR_MIU)

// Exact f32 roundings of k/10 (matches jnp.arange(11)/10 edges). Using decimal
// literals is essential: e.g. 9*0.1f rounds UP past fl(0.9) and would flip bins.
__device__ __forceinline__ float ghmr_edge(int k) {
    const float e[9] = {0.1f, 0.2f, 0.3f, 0.4f, 0.5f, 0.6f, 0.7f, 0.8f, 0.9f};
    return e[k];
}

// ---------------------------------------------------------------------------
// Kernel 0: zero the scratch accumulators (graph-safe re-init every launch).
// d_ws layout:  [0..39]  u32 ge[10]   (ge[0]=#valid, ge[k]=#(g>=k/10), k=1..9)
//               [40..43] f32 wsum
//               [64..107] f32 beta[10] then f32 total at [104]? -> beta[10]+total = 11 floats
__global__ void ghmr_init(unsigned* __restrict__ g_ge, float* __restrict__ g_wsum) {
    if (threadIdx.x < GHMR_BINS) g_ge[threadIdx.x] = 0u;
    if (threadIdx.x == GHMR_BINS) *g_wsum = 0.0f;
}

// ---------------------------------------------------------------------------
// Kernel 1: cumulative histogram. Register counters -> wave32 shuffle tree ->
// one LDS atomic per counter per wave -> one global atomic per counter per block.
__global__ void ghmr_hist(const float* __restrict__ tgt,
                          const float* __restrict__ dlt,
                          const float* __restrict__ wgt,
                          unsigned* __restrict__ g_ge,
                          float* __restrict__ g_wsum,
                          int n) {
    __shared__ unsigned s_ge[GHMR_BINS];
    __shared__ float s_ws;
    if (threadIdx.x < GHMR_BINS) s_ge[threadIdx.x] = 0u;
    if (threadIdx.x == GHMR_BINS) s_ws = 0.0f;
    __syncthreads();

    unsigned ge[GHMR_BINS];
#pragma unroll
    for (int k = 0; k < GHMR_BINS; ++k) ge[k] = 0u;
    float ws = 0.0f;

    auto acc = [&](float t, float d, float w) {
        if (w > 0.0f) {
            float diff  = __fsub_rn(d, t);
            float denom = __fsqrt_rn(__fadd_rn(__fmul_rn(diff, diff), GHMR_MIU2));
            float g     = __fdiv_rn(fabsf(diff), denom);
            ge[0] += 1u;
#pragma unroll
            for (int k = 1; k < GHMR_BINS; ++k) ge[k] += (g >= ghmr_edge(k - 1)) ? 1u : 0u;
            ws = __fadd_rn(ws, w);
        }
    };

    const int n4     = n >> 2;
    const int stride = (int)(gridDim.x * blockDim.x);
    const float4* t4 = (const float4*)tgt;
    const float4* d4 = (const float4*)dlt;
    const float4* w4 = (const float4*)wgt;

    for (int i = (int)(blockIdx.x * blockDim.x + threadIdx.x); i < n4; i += stride) {
        if (i + stride < n4) {  // gfx1250: global_prefetch_b8, prime next tile
            __builtin_prefetch(&t4[i + stride], 0, 1);
            __builtin_prefetch(&d4[i + stride], 0, 1);
            __builtin_prefetch(&w4[i + stride], 0, 1);
        }
        float4 t = t4[i], d = d4[i], w = w4[i];
        acc(t.x, d.x, w.x); acc(t.y, d.y, w.y);
        acc(t.z, d.z, w.z); acc(t.w, d.w, w.w);
    }
    // scalar tail (n % 4)
    for (int i = (n4 << 2) + (int)(blockIdx.x * blockDim.x + threadIdx.x); i < n; i += stride)
        acc(tgt[i], dlt[i], wgt[i]);

    // wave32 shuffle reduction (deterministic fixed tree), then LDS atomics
    const int lane = (int)(threadIdx.x & 31u);
#pragma unroll
    for (int k = 0; k < GHMR_BINS; ++k) {
        unsigned v = ge[k];
        for (int off = 16; off > 0; off >>= 1) v += __shfl_down(v, off, 32);
        if (lane == 0 && v) atomicAdd(&s_ge[k], v);
    }
    {
        float v = ws;
        for (int off = 16; off > 0; off >>= 1) v += __shfl_down(v, off, 32);
        if (lane == 0) atomicAdd(&s_ws, v);
    }
    __syncthreads();
    if (threadIdx.x < GHMR_BINS) {
        unsigned v = s_ge[threadIdx.x];
        if (v) atomicAdd(&g_ge[threadIdx.x], v);
    }
    if (threadIdx.x == GHMR_BINS) atomicAdd(g_wsum, s_ws);
}

// ---------------------------------------------------------------------------
// Kernel 2: finalize. counts[b] = ge[b] - ge[b+1]; beta[b] follows reference
// rounding order exactly: beta = (total/max(c,1)) / max(n_nonempty,1).
// out later = (asl1 * beta) / total  -> store total alongside beta.
__global__ void ghmr_finalize(const unsigned* __restrict__ g_ge,
                              const float* __restrict__ g_wsum,
                              float* __restrict__ beta_out /* 11 floats: beta[10], total */) {
    if (threadIdx.x != 0) return;
    float total = *g_wsum;
    if (total < 1.0f) total = 1.0f;

    unsigned ge[GHMR_BINS + 1];
#pragma unroll
    for (int k = 0; k < GHMR_BINS; ++k) ge[k] = g_ge[k];
    ge[GHMR_BINS] = 0u;

    float counts[GHMR_BINS];
    int n_ne = 0;
#pragma unroll
    for (int b = 0; b < GHMR_BINS; ++b) {
        counts[b] = (float)(ge[b] - ge[b + 1]);
        if (counts[b] > 0.0f) ++n_ne;
    }
    float nn = (float)n_ne;
    if (nn < 1.0f) nn = 1.0f;

#pragma unroll
    for (int b = 0; b < GHMR_BINS; ++b) {
        float pbw = (counts[b] > 0.0f)
                        ? __fdiv_rn(total, fmaxf(counts[b], 1.0f))
                        : 0.0f;
        beta_out[b] = __fdiv_rn(pbw, nn);
    }
    beta_out[GHMR_BINS] = total;
}

// ---------------------------------------------------------------------------
// Kernel 3: apply. Recompute asl1/g (inputs are L2-resident after pass 1:
// 88 MB < 192 MB L2), gather beta[bin] from LDS broadcast, write output.
__global__ void ghmr_apply(const float* __restrict__ tgt,
                           const float* __restrict__ dlt,
                           const float* __restrict__ wgt,
                           const float* __restrict__ beta_in,
                           float* __restrict__ out,
                           int n) {
    __shared__ float s_beta[GHMR_BINS + 1];
    if (threadIdx.x < GHMR_BINS + 1) s_beta[threadIdx.x] = beta_in[threadIdx.x];
    __syncthreads();
    const float total = s_beta[GHMR_BINS];

    auto one = [&](float t, float d, float w) -> float {
        float diff  = __fsub_rn(d, t);
        float denom = __fsqrt_rn(__fadd_rn(__fmul_rn(diff, diff), GHMR_MIU2));
        float asl1  = __fsub_rn(denom, GHMR_MIU);
        if (w > 0.0f) {
            float g = __fdiv_rn(fabsf(diff), denom);
            int bin = 0;
#pragma unroll
            for (int k = 0; k < GHMR_BINS - 1; ++k) bin += (g >= ghmr_edge(k)) ? 1 : 0;
            return __fdiv_rn(__fmul_rn(asl1, s_beta[bin]), total);
        }
        return 0.0f;  // beta==0 -> asl1*0/total == +0
    };

    const int n4     = n >> 2;
    const int stride = (int)(gridDim.x * blockDim.x);
    const float4* t4 = (const float4*)tgt;
    const float4* d4 = (const float4*)dlt;
    const float4* w4 = (const float4*)wgt;
    float4* o4       = (float4*)out;

    for (int i = (int)(blockIdx.x * blockDim.x + threadIdx.x); i < n4; i += stride) {
        if (i + stride < n4) {
            __builtin_prefetch(&t4[i + stride], 0, 1);
            __builtin_prefetch(&d4[i + stride], 0, 1);
            __builtin_prefetch(&w4[i + stride], 0, 1);
        }
        float4 t = t4[i], d = d4[i], w = w4[i];
        float4 o;
        o.x = one(t.x, d.x, w.x);
        o.y = one(t.y, d.y, w.y);
        o.z = one(t.z, d.z, w.z);
        o.w = one(t.w, d.w, w.w);
        o4[i] = o;
    }
    for (int i = (n4 << 2) + (int)(blockIdx.x * blockDim.x + threadIdx.x); i < n; i += stride)
        out[i] = one(tgt[i], dlt[i], wgt[i]);
}

// ---------------------------------------------------------------------------
extern "C" void kernel_launch(void* const* d_in, const int* in_sizes, int n_in,
                              void* d_out, int out_size, void* d_ws, size_t ws_size,
                              hipStream_t stream) {
    const float* tgt = (const float*)d_in[0];  // delta_targets
    const float* dlt = (const float*)d_in[1];  // deltas
    const float* wgt = (const float*)d_in[2];  // delta_weights
    float* out = (float*)d_out;
    const int n = in_sizes[0];

    unsigned* g_ge = (unsigned*)d_ws;                       // 10 x u32
    float* g_wsum  = (float*)((char*)d_ws + 40);            // 1 x f32
    float* g_beta  = (float*)((char*)d_ws + 64);            // 11 x f32 (beta[10], total)

    const int threads = 256;                                 // 8 wave32s / block
    int blocks = (n / 4 + threads - 1) / threads;
    if (blocks > 2048) blocks = 2048;
    if (blocks < 1) blocks = 1;

    ghmr_init<<<1, 32, 0, stream>>>(g_ge, g_wsum);
    ghmr_hist<<<blocks, threads, 0, stream>>>(tgt, dlt, wgt, g_ge, g_wsum, n);
    ghmr_finalize<<<1, 32, 0, stream>>>(g_ge, g_wsum, g_beta);
    ghmr_apply<<<blocks, threads, 0, stream>>>(tgt, dlt, wgt, g_beta, out, n);
}